// EssentialMatrixEstimator_81741817578204
// MI455X (gfx1250) — compile-verified
//
#include <hip/hip_runtime.h>
#include <math.h>

typedef __attribute__((ext_vector_type(2))) float v2f;
typedef __attribute__((ext_vector_type(8))) float v8f;

#define NP   3072
#define LDP  3073
#define NWAVES 1024
#define ROWS_PER_WAVE 3
#define EPSF 1e-8f

// ---------------------------------------------------------------- helpers
__device__ __forceinline__ void ins3(float v, float& m1, float& m2, float& m3) {
    if (v > m3) {
        if (v > m1)      { m3 = m2; m2 = m1; m1 = v; }
        else if (v > m2) { m3 = m2; m2 = v; }
        else             { m3 = v; }
    }
}

__device__ __forceinline__ float det3(const float M[3][3]) {
    return M[0][0]*(M[1][1]*M[2][2]-M[1][2]*M[2][1])
         - M[0][1]*(M[1][0]*M[2][2]-M[1][2]*M[2][0])
         + M[0][2]*(M[1][0]*M[2][1]-M[1][1]*M[2][0]);
}
__device__ __forceinline__ float signf(float x) { return (x > 0.f) ? 1.f : ((x < 0.f) ? -1.f : 0.f); }

// ---------------------------------------------------------------- K0: K_inv
__global__ void k0_kinv(const float* __restrict__ K, float* __restrict__ kinv) {
    if (threadIdx.x == 0 && blockIdx.x == 0) {
        float a=K[0],b=K[1],c=K[2],d=K[3],e=K[4],f=K[5],g=K[6],h=K[7],i=K[8];
        float A =  e*i-f*h, B = -(d*i-f*g), C = d*h-e*g;
        float det = a*A + b*B + c*C;
        float inv = 1.0f/det;
        kinv[0]= A*inv;          kinv[1]=-(b*i-c*h)*inv; kinv[2]= (b*f-c*e)*inv;
        kinv[3]= B*inv;          kinv[4]= (a*i-c*g)*inv; kinv[5]=-(a*f-c*d)*inv;
        kinv[6]= C*inv;          kinv[7]=-(a*h-b*g)*inv; kinv[8]= (a*e-b*d)*inv;
    }
}

// ---------------------------------------------------------------- K1: row top-3 (one wave per row)
__global__ __launch_bounds__(256) void k1_rowtop3(const float* __restrict__ P, float* __restrict__ tr) {
    int wave = (blockIdx.x * blockDim.x + threadIdx.x) >> 5;
    int lane = threadIdx.x & 31;
    if (wave >= NP) return;
    const float* row = P + (size_t)wave * LDP;
    float m1=-1e30f, m2=-1e30f, m3=-1e30f;
    for (int j = lane; j < NP; j += 32) ins3(row[j], m1, m2, m3);
    for (int off = 16; off >= 1; off >>= 1) {
        float o1 = __shfl_xor(m1, off, 32);
        float o2 = __shfl_xor(m2, off, 32);
        float o3 = __shfl_xor(m3, off, 32);
        ins3(o1, m1, m2, m3); ins3(o2, m1, m2, m3); ins3(o3, m1, m2, m3);
    }
    if (lane == 0) tr[wave] = m3;
}

// ---------------------------------------------------------------- K2: col top-3 (thread per column, coalesced)
__global__ __launch_bounds__(256) void k2_coltop3(const float* __restrict__ P, float* __restrict__ tc) {
    int j = blockIdx.x * blockDim.x + threadIdx.x;
    if (j >= NP) return;
    float m1=-1e30f, m2=-1e30f, m3=-1e30f;
    for (int i = 0; i < NP; ++i) ins3(P[(size_t)i * LDP + j], m1, m2, m3);
    tc[j] = m3;
}

// ---------------------------------------------------------------- K3: fused mask + raw 9x9 moments via WMMA
// M_raw(16x16 pad) += A(16x4) x B(4x16):  A[m][k]=w_k*a_k[m], B[k][n]=a_k[n]
// ISA A layout 16x4 f32: VGPR0 lanes0-15 K=0 / lanes16-31 K=2 ; VGPR1 K=1 / K=3.
// B 4x16 mirrored (VGPR0: K=0|K=2, VGPR1: K=1|K=3).
__global__ __launch_bounds__(256) void k3_accum(const float* __restrict__ P,
                                               const float* __restrict__ kinv,
                                               const float* __restrict__ tr,
                                               const float* __restrict__ tc,
                                               float* __restrict__ partials) {
    __shared__ float tcs[NP];
    __shared__ float h2full[3 * NP];   // [0]:x2  [1]:y2  [2]:1.0f  — kills the per-lane select

    int tid  = threadIdx.x;
    int wave = (blockIdx.x * blockDim.x + tid) >> 5;
    int lane = tid & 31;
    int half = lane >> 4;           // 0: K pairs 0,1   1: K pairs 2,3
    int m    = lane & 15;           // output row index this lane builds
    float ki00=kinv[0], ki01=kinv[1], ki02=kinv[2];
    float ki10=kinv[3], ki11=kinv[4], ki12=kinv[5];

    // stage per-column tables into LDS (block-wide), including constant-1 plane
    for (int j = tid; j < NP; j += 256) {
        float px = (float)(j & 63), py = (float)(j >> 6);
        tcs[j]            = tc[j];
        h2full[j]         = ki00*px + ki01*py + ki02;   // x2
        h2full[NP + j]    = ki10*px + ki11*py + ki12;   // y2
        h2full[2*NP + j]  = 1.0f;
    }
    __syncthreads();

    int r  = m / 3;                 // h1 component selector (m<9)
    int cc = m - 3*r;               // h2 component selector
    const float* htab = h2full + cc * NP;   // unconditional per-lane table

    v8f acc0 = {};
    v8f acc1 = {};

    for (int rr = 0; rr < ROWS_PER_WAVE; ++rr) {
        int i = wave + rr * NWAVES;
        if (i >= NP) break;                          // uniform per wave
        float px1 = (float)(i & 63), py1 = (float)(i >> 6);
        float x1 = ki00*px1 + ki01*py1 + ki02;
        float y1 = ki10*px1 + ki11*py1 + ki12;
        float h1v = (m < 9) ? (r==0 ? x1 : (r==1 ? y1 : 1.0f)) : 0.0f;
        float tri = tr[i];
        const float* prow = P + (size_t)i * LDP;

        for (int jb = 0; jb < NP; jb += 8) {
            // ---- WMMA #0 : columns jb..jb+3 ----
            int jA0 = jb + 2*half;
            float pA0 = prow[jA0], pB0 = prow[jA0+1];
            float tA0 = tcs[jA0],  tB0 = tcs[jA0+1];
            float hA0 = htab[jA0], hB0 = htab[jA0+1];
            float wA0 = ((pA0 >= tri) & (pA0 >= tA0) & (pA0 > 0.01f)) ? pA0 : 0.0f;
            float wB0 = ((pB0 >= tri) & (pB0 >= tB0) & (pB0 > 0.01f)) ? pB0 : 0.0f;
            float aA0 = h1v * hA0;
            float aB0 = h1v * hB0;
            v2f Aop0, Bop0;
            Aop0[0] = wA0 * aA0;  Aop0[1] = wB0 * aB0;
            Bop0[0] = aA0;        Bop0[1] = aB0;
            acc0 = __builtin_amdgcn_wmma_f32_16x16x4_f32(
                       false, Aop0, false, Bop0, (short)0, acc0, false, false);

            // ---- WMMA #1 : columns jb+4..jb+7 ----
            int jA1 = jb + 4 + 2*half;
            float pA1 = prow[jA1], pB1 = prow[jA1+1];
            float tA1 = tcs[jA1],  tB1 = tcs[jA1+1];
            float hA1 = htab[jA1], hB1 = htab[jA1+1];
            float wA1 = ((pA1 >= tri) & (pA1 >= tA1) & (pA1 > 0.01f)) ? pA1 : 0.0f;
            float wB1 = ((pB1 >= tri) & (pB1 >= tB1) & (pB1 > 0.01f)) ? pB1 : 0.0f;
            float aA1 = h1v * hA1;
            float aB1 = h1v * hB1;
            v2f Aop1, Bop1;
            Aop1[0] = wA1 * aA1;  Aop1[1] = wB1 * aB1;
            Bop1[0] = aA1;        Bop1[1] = aB1;
            acc1 = __builtin_amdgcn_wmma_f32_16x16x4_f32(
                       false, Aop1, false, Bop1, (short)0, acc1, false, false);
        }
    }
    // deterministic merge of the two accumulators, dump 16x16 partial
    float* pout = partials + (size_t)wave * 256;
    #pragma unroll
    for (int v = 0; v < 8; ++v) pout[v*32 + lane] = acc0[v] + acc1[v];
}

// ---------------------------------------------------------------- K4: deterministic reduce + serial solve
__global__ __launch_bounds__(256) void k4_solve(const float* __restrict__ partials,
                                                float* __restrict__ out) {
    __shared__ float Msum[256];
    int t = threadIdx.x;
    float acc = 0.f;
    for (int w = 0; w < NWAVES; ++w) acc += partials[(size_t)w * 256 + t];
    Msum[t] = acc;
    __syncthreads();
    if (t != 0) return;

    // unpack raw 9x9 from WMMA C/D layout: (M,N): M<8 -> Msum[M*32+N]; M>=8 -> Msum[(M-8)*32+16+N]
    float Mr[9][9];
    for (int rr = 0; rr < 9; ++rr)
        for (int c2 = 0; c2 < 9; ++c2)
            Mr[rr][c2] = Msum[(rr & 7)*32 + ((rr >= 8) ? 16 : 0) + c2];

    // Hartley moments are entries of the raw Gram matrix (a = [x1x2,x1y2,x1, y1x2,y1y2,y1, x2,y2,1]):
    float sw  = Mr[8][8];
    float sx1 = Mr[2][8], sy1 = Mr[5][8], sq1 = Mr[2][2] + Mr[5][5];
    float sx2 = Mr[8][6], sy2 = Mr[8][7], sq2 = Mr[6][6] + Mr[7][7];

    const float SQ2 = 1.4142135623730951f;
    const float IV3 = 0.5773502691896258f;
    float wsum = sw + EPSF;
    float c1x = sx1/wsum, c1y = sy1/wsum;
    float var1 = (sq1 - 2.f*(c1x*sx1 + c1y*sy1) + (c1x*c1x + c1y*c1y)*sw)/wsum;
    float sc1 = SQ2 / (sqrtf(var1 + EPSF) + EPSF);
    float c2x = sx2/wsum, c2y = sy2/wsum;
    float var2 = (sq2 - 2.f*(c2x*sx2 + c2y*sy2) + (c2x*c2x + c2y*c2y)*sw)/wsum;
    float sc2 = SQ2 / (sqrtf(var2 + EPSF) + EPSF);

    float T1m[3][3] = {{sc1,0.f,-sc1*c1x},{0.f,sc1,-sc1*c1y},{0.f,0.f,1.f}};
    float T2m[3][3] = {{sc2,0.f,-sc2*c2x},{0.f,sc2,-sc2*c2y},{0.f,0.f,1.f}};

    // normalization as similarity on 9x9: Mp = G Mr G^T, G = kron(T1,T2)
    float G[9][9];
    for (int a=0;a<3;++a) for (int b=0;b<3;++b)
        for (int a2=0;a2<3;++a2) for (int b2=0;b2<3;++b2)
            G[3*a+b][3*a2+b2] = T1m[a][a2]*T2m[b][b2];
    float GM[9][9], Mp[9][9];
    for (int a=0;a<9;++a) for (int b=0;b<9;++b) {
        float s=0.f; for (int k=0;k<9;++k) s += G[a][k]*Mr[k][b]; GM[a][b]=s;
    }
    for (int a=0;a<9;++a) for (int b=0;b<9;++b) {
        float s=0.f; for (int k=0;k<9;++k) s += GM[a][k]*G[b][k]; Mp[a][b]=s;
    }
    float lam = 0.f; for (int k=0;k<9;++k) lam += Mp[k][k];
    float Ms9[9][9];
    for (int a=0;a<9;++a) for (int b=0;b<9;++b) Ms9[a][b] = ((a==b)?lam:0.f) - Mp[a][b];

    float v9[9]; for (int k=0;k<9;++k) v9[k] = 1.0f/3.0f;
    for (int it=0; it<50; ++it) {
        float nv[9], nrm=0.f;
        for (int a=0;a<9;++a) { float s=0.f; for (int b=0;b<9;++b) s += Ms9[a][b]*v9[b]; nv[a]=s; nrm+=s*s; }
        nrm = sqrtf(nrm);
        for (int a=0;a<9;++a) v9[a] = nv[a]/(nrm+EPSF);
    }
    float Er[3][3] = {{v9[0],v9[1],v9[2]},{v9[3],v9[4],v9[5]},{v9[6],v9[7],v9[8]}};
    // E = T2^T Er T1
    float tm[3][3], E[3][3];
    for (int a=0;a<3;++a) for (int b=0;b<3;++b) {
        float s=0.f; for (int k=0;k<3;++k) s += Er[a][k]*T1m[k][b]; tm[a][b]=s;
    }
    for (int a=0;a<3;++a) for (int b=0;b<3;++b) {
        float s=0.f; for (int k=0;k<3;++k) s += T2m[k][a]*tm[k][b]; E[a][b]=s;
    }

    // ---- project_E ----
    float B3[3][3];
    for (int a=0;a<3;++a) for (int b=0;b<3;++b) {
        float s=0.f; for (int k=0;k<3;++k) s += E[k][a]*E[k][b]; B3[a][b]=s;
    }
    float lamB = B3[0][0]+B3[1][1]+B3[2][2];
    float v1v[3] = {IV3,IV3,IV3};
    for (int it=0; it<50; ++it) {
        float nv[3], nrm=0.f;
        for (int a=0;a<3;++a){ float s=0.f; for (int b=0;b<3;++b) s+=B3[a][b]*v1v[b]; nv[a]=s; nrm+=s*s; }
        nrm = sqrtf(nrm);
        for (int a=0;a<3;++a) v1v[a]=nv[a]/(nrm+EPSF);
    }
    float A2[3][3];
    for (int a=0;a<3;++a) for (int b=0;b<3;++b) A2[a][b] = ((a==b)?lamB:0.f) - B3[a][b];
    float v3v[3] = {IV3,IV3,IV3};
    for (int it=0; it<50; ++it) {
        float nv[3], nrm=0.f;
        for (int a=0;a<3;++a){ float s=0.f; for (int b=0;b<3;++b) s+=A2[a][b]*v3v[b]; nv[a]=s; nrm+=s*s; }
        nrm = sqrtf(nrm);
        for (int a=0;a<3;++a) v3v[a]=nv[a]/(nrm+EPSF);
    }
    float v2v[3] = { v3v[1]*v1v[2]-v3v[2]*v1v[1],
                     v3v[2]*v1v[0]-v3v[0]*v1v[2],
                     v3v[0]*v1v[1]-v3v[1]*v1v[0] };
    float n2 = sqrtf(v2v[0]*v2v[0]+v2v[1]*v2v[1]+v2v[2]*v2v[2]);
    for (int a=0;a<3;++a) v2v[a] /= (n2+EPSF);

    float V[3][3];
    for (int a=0;a<3;++a){ V[a][0]=v1v[a]; V[a][1]=v2v[a]; V[a][2]=v3v[a]; }
    float sV = signf(det3(V));
    for (int a=0;a<3;++a) V[a][2] *= sV;

    float Ev1[3], Ev2[3];
    for (int a=0;a<3;++a){
        Ev1[a] = E[a][0]*V[0][0]+E[a][1]*V[1][0]+E[a][2]*V[2][0];
        Ev2[a] = E[a][0]*V[0][1]+E[a][1]*V[1][1]+E[a][2]*V[2][1];
    }
    float n_1 = sqrtf(Ev1[0]*Ev1[0]+Ev1[1]*Ev1[1]+Ev1[2]*Ev1[2]);
    float n_2 = sqrtf(Ev2[0]*Ev2[0]+Ev2[1]*Ev2[1]+Ev2[2]*Ev2[2]);
    float sa = 0.5f*(n_1+n_2);
    float u1[3], u2[3];
    for (int a=0;a<3;++a){ u1[a]=Ev1[a]/(n_1+EPSF); u2[a]=Ev2[a]/(n_2+EPSF); }
    float u3[3] = { u1[1]*u2[2]-u1[2]*u2[1], u1[2]*u2[0]-u1[0]*u2[2], u1[0]*u2[1]-u1[1]*u2[0] };
    float U[3][3];
    for (int a=0;a<3;++a){ U[a][0]=u1[a]; U[a][1]=u2[a]; U[a][2]=u3[a]; }
    float sU = signf(det3(U));
    for (int a=0;a<3;++a) U[a][2] *= sU;     // killed by the zero singular value anyway

    for (int a=0;a<3;++a)
        for (int b=0;b<3;++b)
            out[3*a+b] = sa*(U[a][0]*V[b][0] + U[a][1]*V[b][1]);
}

// ---------------------------------------------------------------- launch
extern "C" void kernel_launch(void* const* d_in, const int* in_sizes, int n_in,
                              void* d_out, int out_size, void* d_ws, size_t ws_size,
                              hipStream_t stream) {
    const float* P = (const float*)d_in[0];
    const float* K = (const float*)d_in[1];
    float* ws       = (float*)d_ws;
    float* kinv     = ws;                         // 16
    float* tr       = ws + 16;                    // 3072
    float* tc       = ws + 16 + NP;               // 3072
    float* partials = ws + 16 + 2*NP;             // NWAVES*256  (~1 MB)
    float* out = (float*)d_out;

    k0_kinv   <<<1, 32, 0, stream>>>(K, kinv);
    k1_rowtop3<<<NP/8, 256, 0, stream>>>(P, tr);      // one wave per row
    k2_coltop3<<<NP/256, 256, 0, stream>>>(P, tc);
    k3_accum  <<<NWAVES/8, 256, 0, stream>>>(P, kinv, tr, tc, partials);
    k4_solve  <<<1, 256, 0, stream>>>(partials, out);
}